// LightGCN_80900003988292
// MI455X (gfx1250) — compile-verified
//
#include <hip/hip_runtime.h>
#include <stdint.h>

// LightGCN forward for MI455X (gfx1250).
// Edge streaming uses CDNA5 async global->LDS copies (ASYNCcnt) with double
// buffering and NT temporal hint (edge data is stream-once-per-layer; keep the
// 192MB L2 for the 38.4MB embedding tables which every gather/atomic hits).
// Scatter uses hardware f32 global atomics (the binding resource).
// spmm_kernel is defined FIRST so the disasm snippet shows its codegen.

#define N_USERS   100000
#define N_ITEMS   50000
#define N_NODES   150000
#define EMB       64
#define N_LAYERS  3
#define BATCH     16384
#define ALPHA_F   0.25f

#define EPC        1024     // edges per chunk (per block): 3 * 4KB LDS per buffer
#define SPMM_BLOCKS 2048

__device__ __forceinline__ uint32_t lds_addr32(const void* p) {
  // Generic pointer to __shared__: low 32 bits of the flat address are the
  // LDS byte offset (ISA 10.2 aperture mapping).
  return (uint32_t)(uintptr_t)p;
}

__device__ __forceinline__ void async_load_b128_nt(uint32_t lds, uint64_t gaddr) {
  // Non-temporal: edge arrays are streamed once per layer; don't pollute L2.
  asm volatile("global_load_async_to_lds_b128 %0, %1, off th:TH_LOAD_NT"
               :: "v"(lds), "v"(gaddr) : "memory");
}

__global__ void __launch_bounds__(256)
spmm_kernel(const int* __restrict__ rows, const int* __restrict__ cols,
            const float* __restrict__ vals, const float* __restrict__ x,
            float* __restrict__ y, int n_edges) {
  __shared__ int   s_rows[2][EPC];
  __shared__ int   s_cols[2][EPC];
  __shared__ float s_vals[2][EPC];

  const int tid  = threadIdx.x;   // 0..255
  const int wave = tid >> 5;      // 0..7 (wave32)
  const int lane = tid & 31;
  const int nchunks = (n_edges + EPC - 1) / EPC;

  const int c0 = blockIdx.x;
  if (c0 >= nchunks) return;      // uniform per block: safe early-out

  auto issue = [&](int chunk, int buf) {
    // Each thread moves 4 edges (16B) of each array; clamp the tail so the
    // async engine never reads past the 10M-element arrays (n_edges % 4 == 0).
    int g = chunk * EPC + tid * 4;
    if (g > n_edges - 4) g = n_edges - 4;
    async_load_b128_nt(lds_addr32(&s_rows[buf][tid * 4]), (uint64_t)(uintptr_t)(rows + g));
    async_load_b128_nt(lds_addr32(&s_cols[buf][tid * 4]), (uint64_t)(uintptr_t)(cols + g));
    async_load_b128_nt(lds_addr32(&s_vals[buf][tid * 4]), (uint64_t)(uintptr_t)(vals + g));
  };

  issue(c0, 0);
  int buf = 0;
  for (int c = c0; c < nchunks; c += gridDim.x, buf ^= 1) {
    const int cn = c + (int)gridDim.x;
    if (cn < nchunks) {
      issue(cn, buf ^ 1);                                   // prefetch next chunk
      asm volatile("s_wait_asynccnt 0x3" ::: "memory");     // this wave's chunk c done
    } else {
      asm volatile("s_wait_asynccnt 0x0" ::: "memory");
    }
    __syncthreads();   // all waves' slices of chunk c are now in LDS

    const int base = c * EPC;
    int nE = n_edges - base;
    if (nE > EPC) nE = EPC;

    // One edge per wave-iteration: lane handles emb elements {2*lane, 2*lane+1}.
    for (int e = wave; e < nE; e += 8) {
      const int   row = s_rows[buf][e];   // fused ds_load_2addr_stride64 w/ cols
      const int   col = s_cols[buf][e];
      const float v   = s_vals[buf][e];
      const float2 xv = *(const float2*)(x + (col << 6) + (lane << 1));
      float* yp = y + (row << 6) + (lane << 1);
      unsafeAtomicAdd(yp,     v * xv.x);   // want: global_atomic_add_f32 (no return)
      unsafeAtomicAdd(yp + 1, v * xv.y);
    }
    __syncthreads();   // done reading buf before it is refilled next iteration
  }
}

__global__ void __launch_bounds__(256)
init_kernel(const float4* __restrict__ ue, const float4* __restrict__ ie,
            float4* __restrict__ ego, float4* __restrict__ fin, int n4) {
  int i = blockIdx.x * blockDim.x + threadIdx.x;
  if (i >= n4) return;
  float4 v = (i < N_USERS * (EMB / 4)) ? ue[i] : ie[i - N_USERS * (EMB / 4)];
  ego[i] = v;
  fin[i] = make_float4(ALPHA_F * v.x, ALPHA_F * v.y, ALPHA_F * v.z, ALPHA_F * v.w);
}

__global__ void __launch_bounds__(256)
accum_kernel(float4* __restrict__ fin, const float4* __restrict__ nxt, int n4) {
  int i = blockIdx.x * blockDim.x + threadIdx.x;
  if (i >= n4) return;
  float4 a = fin[i];
  float4 b = nxt[i];
  a.x += ALPHA_F * b.x;
  a.y += ALPHA_F * b.y;
  a.z += ALPHA_F * b.z;
  a.w += ALPHA_F * b.w;
  fin[i] = a;
}

__global__ void __launch_bounds__(256)
gather_kernel(const int* __restrict__ users, const int* __restrict__ pos,
              const int* __restrict__ neg, const float4* __restrict__ fin,
              const float4* __restrict__ ue, const float4* __restrict__ ie,
              float4* __restrict__ out) {
  const int g = blockIdx.x * blockDim.x + threadIdx.x;
  if (g >= BATCH * (EMB / 4)) return;
  const int i = g >> 4;        // batch index
  const int c = g & 15;        // float4 column within the 64-float row
  const int u = users[i];
  const int p = pos[i];
  const int n = neg[i];
  const int S = BATCH * (EMB / 4);
  out[0 * S + g] = fin[u * 16 + c];
  out[1 * S + g] = fin[(N_USERS + p) * 16 + c];
  out[2 * S + g] = fin[(N_USERS + n) * 16 + c];
  out[3 * S + g] = ue[u * 16 + c];
  out[4 * S + g] = ie[p * 16 + c];
  out[5 * S + g] = ie[n * 16 + c];
}

extern "C" void kernel_launch(void* const* d_in, const int* in_sizes, int n_in,
                              void* d_out, int out_size, void* d_ws, size_t ws_size,
                              hipStream_t stream) {
  (void)n_in; (void)out_size; (void)ws_size;
  const int*   rows  = (const int*)d_in[0];
  const int*   cols  = (const int*)d_in[1];
  const float* vals  = (const float*)d_in[2];
  const int*   users = (const int*)d_in[3];
  const int*   pos   = (const int*)d_in[4];
  const int*   neg   = (const int*)d_in[5];
  const float* ue    = (const float*)d_in[6];
  const float* ie    = (const float*)d_in[7];
  const int n_edges  = in_sizes[0];

  // Workspace: ego | next | final, each N_NODES*EMB f32 (38.4MB) -> 115.2MB.
  const size_t nodeFloats = (size_t)N_NODES * EMB;
  float* ego = (float*)d_ws;
  float* nxt = ego + nodeFloats;
  float* fin = nxt + nodeFloats;

  const int n4 = N_NODES * (EMB / 4);
  init_kernel<<<(n4 + 255) / 256, 256, 0, stream>>>(
      (const float4*)ue, (const float4*)ie, (float4*)ego, (float4*)fin, n4);

  for (int l = 0; l < N_LAYERS; ++l) {
    hipMemsetAsync(nxt, 0, nodeFloats * sizeof(float), stream);
    spmm_kernel<<<SPMM_BLOCKS, 256, 0, stream>>>(rows, cols, vals, ego, nxt, n_edges);
    accum_kernel<<<(n4 + 255) / 256, 256, 0, stream>>>(
        (float4*)fin, (const float4*)nxt, n4);
    float* t = ego; ego = nxt; nxt = t;
  }

  gather_kernel<<<(BATCH * (EMB / 4) + 255) / 256, 256, 0, stream>>>(
      users, pos, neg, (const float4*)fin, (const float4*)ue, (const float4*)ie,
      (float4*)d_out);
}